// Doppler_operatorV2_71090298684011
// MI455X (gfx1250) — compile-verified
//
#include <hip/hip_runtime.h>
#include <stdint.h>

#define Mdim 96
#define Ndim 96
#define BC   32
#define MN   9216
#define WRAPW 1.0f

__device__ __forceinline__ float wrapf(float v) {
  // python-style (v + w) % (2w) - w  (result of % in [0, 2w))
  float tw = 2.0f * WRAPW;
  float t = v + WRAPW;
  float r = t - tw * floorf(t / tw);
  return r - WRAPW;
}

// ---------------- Phase 1: d1y3 / d1x3 ----------------
__global__ void k_diffs(const float* __restrict__ x, const float* __restrict__ DM,
                        const float* __restrict__ DN, float* __restrict__ d1y3,
                        float* __restrict__ d1x3) {
  int i = blockIdx.x * blockDim.x + threadIdx.x;
  if (i >= MN * BC) return;
  int t  = i & 31;          // i = (n2*96+m2)*32 + t
  int rm = i >> 5;
  int m2 = rm % Mdim;
  int n2 = rm / Mdim;
  int sb = n2 & 31;
  int ni = 3 * t + (n2 >> 5);
  const float* xs = x + (size_t)sb * (Mdim * Ndim);
  float accY = 0.f;
  for (int p = 0; p < Mdim; ++p)
    accY += DM[m2 * Mdim + p] * xs[p * Ndim + ni];
  float accX = 0.f;
  const float* xr = xs + m2 * Ndim;
  for (int q = 0; q < Ndim; ++q)
    accX += DN[ni * Ndim + q] * xr[q];
  d1y3[i] = wrapf(accY);
  d1x3[i] = wrapf(accX);
}

// ---------------- Phase 2: pre (32 x 9216, row-major) ----------------
__global__ void k_pre(const float* __restrict__ x, const float* __restrict__ eta,
                      const float* __restrict__ DM, const float* __restrict__ DN,
                      const float* __restrict__ d1y3, const float* __restrict__ d1x3,
                      float* __restrict__ pre) {
  int i = blockIdx.x * blockDim.x + threadIdx.x;
  if (i >= MN * BC) return;
  int r = i % MN;
  int t = i / MN;
  int mm = r % Mdim;
  int nn = r / Mdim;
  float acc = 0.f;
  for (int p = 0; p < Mdim; ++p)
    acc += DM[p * Mdim + mm] * d1y3[(nn * Mdim + p) * BC + t];
  for (int q = 0; q < Ndim; ++q)
    acc += DN[q * Ndim + nn] * d1x3[(q * Mdim + mm) * BC + t];
  int bi = t & 7, ci = t >> 3;
  acc += eta[0] * x[(((size_t)(bi * 4 + ci)) * Mdim + mm) * Ndim + nn];
  pre[(size_t)t * MN + r] = acc;
}

// ---------------- Phase 3: y = pre @ A_w^T via f32 WMMA ----------------
typedef __attribute__((ext_vector_type(2))) float v2f;
typedef __attribute__((ext_vector_type(8))) float v8f;

#define RT     128              // r-chunk per LDS stage
#define RPAD   132              // 132 mod 64 == 4 -> conflict-free b64 reads
#define NCHUNK (MN / RT)        // 72

__device__ __forceinline__ void async_b128(uint32_t ldsoff, const float* g) {
  asm volatile("global_load_async_to_lds_b128 %0, %1, off"
               :: "v"(ldsoff), "v"(g) : "memory");
}

__global__ void __launch_bounds__(32)
k_gemm(const float* __restrict__ pre, const float* __restrict__ Aw,
       float* __restrict__ out) {
  __shared__ float ldsW[2][16 * RPAD];   // A_w tile: 16 cols x 128 r (padded)
  __shared__ float ldsP[2][BC * RPAD];   // pre tile: 32 rows x 128 r (padded)

  const int l  = threadIdx.x;            // wave32, all lanes active
  const int k0 = blockIdx.x * 16;        // this wave owns columns k0..k0+15

  v8f acc0 = {};                         // t = 0..15
  v8f acc1 = {};                         // t = 16..31

  auto fill = [&](int buf, int ch) {
    const int r0 = ch * RT;
    const float* gW = Aw + (size_t)k0 * MN + r0 + 4 * l;
#pragma unroll
    for (int c = 0; c < 16; ++c) {
      uint32_t off = (uint32_t)(uintptr_t)(&ldsW[buf][c * RPAD + 4 * l]);
      async_b128(off, gW + (size_t)c * MN);
    }
    const float* gP = pre + r0 + 4 * l;
#pragma unroll
    for (int tt = 0; tt < BC; ++tt) {
      uint32_t off = (uint32_t)(uintptr_t)(&ldsP[buf][tt * RPAD + 4 * l]);
      async_b128(off, gP + (size_t)tt * MN);
    }
  };

  const int row = l & 15;                // C: N = lane&15 ; A/B K-half select:
  const int kk  = (l >> 4) << 1;         // lanes 0-15 -> K 0,1 ; 16-31 -> K 2,3

  fill(0, 0);
  for (int ch = 0; ch < NCHUNK; ++ch) {
    const int buf = ch & 1;
    if (ch + 1 < NCHUNK) {
      fill(buf ^ 1, ch + 1);
      asm volatile("s_wait_asynccnt 48" ::: "memory");  // chunk `ch` landed
    } else {
      asm volatile("s_wait_asynccnt 0" ::: "memory");
    }
    const float* Wb = &ldsW[buf][0];
    const float* Pb = &ldsP[buf][0];
#pragma unroll 8
    for (int rr = 0; rr < RT; rr += 4) {
      v2f b  = *(const v2f*)(Wb + row * RPAD + rr + kk);          // B 4x16
      v2f a0 = *(const v2f*)(Pb + row * RPAD + rr + kk);          // A 16x4 (t 0-15)
      v2f a1 = *(const v2f*)(Pb + (row + 16) * RPAD + rr + kk);   // A 16x4 (t 16-31)
      acc0 = __builtin_amdgcn_wmma_f32_16x16x4_f32(false, a0, false, b,
                                                   (short)0, acc0, false, false);
      acc1 = __builtin_amdgcn_wmma_f32_16x16x4_f32(false, a1, false, b,
                                                   (short)0, acc1, false, false);
    }
  }

  // C/D layout: lane l -> N = l&15, VGPR v -> M = v + 8*(l>>4).
  // out[bi,ci,mi,ni] = y[bi+8ci, mi+96ni]
  const int k  = k0 + row;
  const int mi = k % 96, ni = k / 96;
  const int half = (l >> 4) * 8;
#pragma unroll
  for (int v = 0; v < 8; ++v) {
    int t0 = v + half;
    int bi0 = t0 & 7, ci0 = t0 >> 3;
    out[(((size_t)(bi0 * 4 + ci0)) * 96 + mi) * 96 + ni] = acc0[v];
    int t1 = t0 + 16;
    int bi1 = t1 & 7, ci1 = t1 >> 3;
    out[(((size_t)(bi1 * 4 + ci1)) * 96 + mi) * 96 + ni] = acc1[v];
  }
}

extern "C" void kernel_launch(void* const* d_in, const int* in_sizes, int n_in,
                              void* d_out, int out_size, void* d_ws, size_t ws_size,
                              hipStream_t stream) {
  (void)in_sizes; (void)n_in; (void)out_size; (void)ws_size;
  const float* x   = (const float*)d_in[0];
  const float* eta = (const float*)d_in[1];
  const float* Aw  = (const float*)d_in[2];
  const float* DM  = (const float*)d_in[3];
  const float* DN  = (const float*)d_in[4];
  float* out = (float*)d_out;

  float* d1y3 = (float*)d_ws;            // 294912 f32
  float* d1x3 = d1y3 + MN * BC;          // 294912 f32
  float* pre  = d1x3 + MN * BC;          // 294912 f32  (~3.5 MB total)

  const int tot = MN * BC;
  k_diffs<<<(tot + 255) / 256, 256, 0, stream>>>(x, DM, DN, d1y3, d1x3);
  k_pre  <<<(tot + 255) / 256, 256, 0, stream>>>(x, eta, DM, DN, d1y3, d1x3, pre);
  k_gemm <<<MN / 16, 32, 0, stream>>>(pre, Aw, out);
}